// Loss_5978594476648
// MI455X (gfx1250) — compile-verified
//
#include <hip/hip_runtime.h>
#include <math.h>
#include <stdint.h>

#define BLOCK 256
#define WAVE  32
#define NWAVE (BLOCK / WAVE)
#define DEPTH 4                      // async pipeline depth (power of 2)
#define TILE_FLOATS (BLOCK * 4)      // floats per stage (each lane: one float4)

typedef int v4i __attribute__((vector_size(16)));
typedef __attribute__((address_space(1))) v4i* g_v4i_ptr;   // global (AS1)
typedef __attribute__((address_space(3))) v4i* l_v4i_ptr;   // LDS (AS3)

// ---- CDNA5 async global->LDS copy (ASYNCcnt path), with asm fallback ----
__device__ __forceinline__ void async_copy_b128(const float4* gsrc, float* lds_dst) {
#if __has_builtin(__builtin_amdgcn_global_load_async_to_lds_b128)
    __builtin_amdgcn_global_load_async_to_lds_b128(
        (g_v4i_ptr)(uintptr_t)gsrc,
        (l_v4i_ptr)(uint32_t)(uintptr_t)lds_dst,   // low 32 bits of generic = LDS offset
        0, 0);
#else
    unsigned int       lofs  = (unsigned int)(uintptr_t)lds_dst;
    unsigned long long gaddr = (unsigned long long)(uintptr_t)gsrc;
    asm volatile("global_load_async_to_lds_b128 %0, %1, off"
                 :: "v"(lofs), "v"(gaddr) : "memory");
#endif
}

#if __has_builtin(__builtin_amdgcn_s_wait_asynccnt)
#define WAIT_ASYNC(n) do { __builtin_amdgcn_s_wait_asynccnt(n); asm volatile("" ::: "memory"); } while (0)
#else
#define WAIT_ASYNC(n) asm volatile("s_wait_asynccnt %0" :: "i"(n) : "memory")
#endif

// Combine two online-softmax partial states (m,s) <- (m,s) (+) (om,os)
__device__ __forceinline__ void online_combine(float& m, float& s, float om, float os) {
    float M = fmaxf(m, om);
    s = s * __expf(m - M) + os * __expf(om - M);
    m = M;
}

// Fold 4 elements directly into the running (m,s) state: 5 exps per 4 elems
// (vs 6 for group-local + combine), and one fewer exp on the carried chain.
__device__ __forceinline__ void accum4(float& m, float& s, float4 x) {
    float gm = fmaxf(fmaxf(x.x, x.y), fmaxf(x.z, x.w));
    float M  = fmaxf(m, gm);
    s = s * __expf(m - M)
      + __expf(x.x - M) + __expf(x.y - M)
      + __expf(x.z - M) + __expf(x.w - M);
    m = M;
}

// One workgroup per row: single-pass online softmax over V logits streamed
// through LDS via async copies, then focal loss at the target index.
__global__ void focal_row_kernel(const float* __restrict__ logits,
                                 const int* __restrict__ target,
                                 float* __restrict__ row_loss,
                                 int V) {
    __shared__ float stage[DEPTH][TILE_FLOATS];

    const int row = blockIdx.x;
    const int tid = threadIdx.x;
    const long long base = (long long)row * (long long)V;
    const float* __restrict__ rp = logits + base;

    // Align vector body to 16B: V is odd, so row bases rotate mod 4 elements.
    const int o = (int)((4 - (base & 3)) & 3);

    float m = -INFINITY;
    float s = 0.0f;

    // Scalar prologue (at most 3 elements)
    if (tid < o) {
        m = rp[tid];
        s = 1.0f;
    }

    const int nvec = (V - o) >> 2;                      // float4 groups
    const float4* __restrict__ vp = (const float4*)(rp + o);
    const int nfull  = nvec / BLOCK;                    // full 256-lane stages
    const int steady = (nfull > DEPTH) ? (nfull - DEPTH) : 0;
    const int pre    = (nfull < DEPTH) ? nfull : DEPTH;

    // Fill the pipeline
    for (int k = 0; k < pre; ++k)
        async_copy_b128(vp + (size_t)k * BLOCK + tid, &stage[k & (DEPTH - 1)][tid * 4]);

    // Steady state: wait for oldest stage, consume from LDS, refill its slot.
    // Each lane reads back only the 16B it copied, so per-wave ASYNCcnt
    // ordering suffices and no workgroup barrier is needed.
    for (int k = 0; k < steady; ++k) {
        WAIT_ASYNC(DEPTH - 1);
        float4 x = *(const float4*)&stage[k & (DEPTH - 1)][tid * 4];
        accum4(m, s, x);
        async_copy_b128(vp + (size_t)(k + DEPTH) * BLOCK + tid,
                        &stage[k & (DEPTH - 1)][tid * 4]);
    }

    // Drain
    WAIT_ASYNC(0);
    for (int k = steady; k < nfull; ++k) {
        float4 x = *(const float4*)&stage[k & (DEPTH - 1)][tid * 4];
        accum4(m, s, x);
    }

    // Leftover float4 groups (nvec % BLOCK): direct b128 loads
    for (int g = nfull * BLOCK + tid; g < nvec; g += BLOCK) {
        accum4(m, s, vp[g]);
    }

    // Scalar epilogue (at most 3 elements)
    const int done = o + (nvec << 2);
    for (int i = done + tid; i < V; i += BLOCK) {
        online_combine(m, s, rp[i], 1.0f);
    }

    // Wave32 reduction (butterfly)
    #pragma unroll
    for (int off = WAVE / 2; off > 0; off >>= 1) {
        float om = __shfl_xor(m, off, WAVE);
        float os = __shfl_xor(s, off, WAVE);
        online_combine(m, s, om, os);
    }

    // Cross-wave combine through LDS
    __shared__ float sm[NWAVE];
    __shared__ float ss[NWAVE];
    const int wave = tid / WAVE;
    const int lane = tid % WAVE;
    if (lane == 0) { sm[wave] = m; ss[wave] = s; }
    __syncthreads();

    if (tid == 0) {
        float M = sm[0], S = ss[0];
        #pragma unroll
        for (int w = 1; w < NWAVE; ++w) online_combine(M, S, sm[w], ss[w]);

        const int tgt = target[row];
        const float xt = rp[tgt];
        const float logpt = xt - M - __logf(S);
        const float pt = __expf(logpt);

        const float u  = 1.0f - pt;
        const float u2 = u * u;
        const float p5 = u2 * u2 * u;  // gamma = 5
        const float p3 = u2 * u;       // gamma = 3
        const float p1 = u;            // gamma = 1
        const float w  = (pt < 0.2f) ? p5 : ((pt < 0.5f) ? p3 : p1);

        row_loss[row] = -w * logpt;
    }
}

// Deterministic final reduction: mean over B row losses -> out[0]
__global__ void reduce_mean_kernel(const float* __restrict__ row_loss,
                                   float* __restrict__ out, int B) {
    __shared__ float buf[BLOCK];
    float s = 0.0f;
    for (int i = threadIdx.x; i < B; i += BLOCK) s += row_loss[i];
    buf[threadIdx.x] = s;
    __syncthreads();
    #pragma unroll
    for (int step = BLOCK / 2; step > 0; step >>= 1) {
        if (threadIdx.x < step) buf[threadIdx.x] += buf[threadIdx.x + step];
        __syncthreads();
    }
    if (threadIdx.x == 0) out[0] = buf[0] / (float)B;
}

extern "C" void kernel_launch(void* const* d_in, const int* in_sizes, int n_in,
                              void* d_out, int out_size, void* d_ws, size_t ws_size,
                              hipStream_t stream) {
    const float* logits = (const float*)d_in[0];
    const int*   target = (const int*)d_in[1];

    const int B = in_sizes[1];            // 2048 rows (size of target)
    const int V = in_sizes[0] / B;        // 50257 vocab

    float* row_loss = (float*)d_ws;       // B floats of scratch

    focal_row_kernel<<<B, BLOCK, 0, stream>>>(logits, target, row_loss, V);
    reduce_mean_kernel<<<1, BLOCK, 0, stream>>>(row_loss, (float*)d_out, B);
}